// PixPro_14525579395641
// MI455X (gfx1250) — compile-verified
//
#include <hip/hip_runtime.h>
#include <math.h>

typedef __attribute__((ext_vector_type(2))) float v2f;
typedef __attribute__((ext_vector_type(8))) float v8f;
typedef __attribute__((ext_vector_type(4))) unsigned int v4u;
typedef __attribute__((ext_vector_type(8))) int v8i;
typedef __attribute__((ext_vector_type(4))) int v4i;

#define NB   8      // batches
#define CC   256    // channels (K dim of GEMM)
#define DD   8
#define HH   16
#define WW   16
#define PP   2048   // D*H*W
#define ROWG 16     // row-tile groups per batch (8 row tiles of 16 each)
#define COLS 8      // column slices per batch
#define CT   16     // column tiles per slice
#define EPSF 1e-6f

// Stage k[n, 0..255, pj..pj+15] (16 KB) into LDS at lds_off via the Tensor
// Data Mover. D# per CDNA5 ISA ch.8: 2D tensor, dim0=PP (contig, stride PP),
// dim1=CC rows; tile 16 x 256, data_size=4B. Rows land contiguously in LDS
// -> layout [c][16] f32.
__device__ __forceinline__ void tdm_stage_ktile(const float* kn, int pj,
                                                unsigned lds_off) {
  unsigned long long ga = (unsigned long long)(size_t)(kn + pj);
  v4u g0 = { 1u,                                   // count=1, no gather
             lds_off,                              // lds_addr
             (unsigned)ga,                         // global_addr[31:0]
             (unsigned)(ga >> 32) | (2u << 30) };  // global_addr[56:32] | type=2
  v8i g1 = { (int)(2u << 16),                           // data_size = 4 bytes
             (int)(((unsigned)PP & 0xFFFFu) << 16),     // tensor_dim0 lo16
             (int)((PP >> 16) | ((CC & 0xFFFF) << 16)), // dim0 hi16 | dim1 lo16
             (int)((CC >> 16) | (16u << 16)),           // dim1 hi16 | tile_dim0=16
             (int)CC,                                   // tile_dim1=256, tile_dim2=0
             (int)PP,                                   // tensor_dim0_stride lo32
             0, 0 };
  v4i z4 = { 0, 0, 0, 0 };
  v8i z8 = { 0, 0, 0, 0, 0, 0, 0, 0 };
  __builtin_amdgcn_tensor_load_to_lds(g0, g1, z4, z4, z8, 0);
}

// Grid: 1024 blocks = NB * ROWG * COLS. Block = 256 threads = 8 waves.
// Wave w of block (n, rowg, cs) computes row-tile rowg*8+w over the CT=16
// column tiles of slice cs, with TDM double-buffered B staging.
__global__ __launch_bounds__(256) void pixpro_main(
    const float* __restrict__ q, const float* __restrict__ kk,
    const float* __restrict__ cq, const float* __restrict__ ck,
    float* __restrict__ partials) {
  const int b     = blockIdx.x;
  const int n     = b >> 7;         // batch
  const int rem   = b & 127;
  const int rowg  = rem >> 3;       // row group 0..15
  const int cs    = rem & 7;        // column slice 0..7
  const int jt0   = cs * CT;        // first column tile of this slice
  const int tid   = threadIdx.x;
  const int wave  = tid >> 5;
  const int lane  = tid & 31;
  const int lhalf = lane >> 4;      // 0: lanes 0-15, 1: lanes 16-31
  const int lmod  = lane & 15;
  const int p0    = (rowg * 8 + wave) * 16;  // first row of this wave's tile

  const float* qn = q  + (size_t)n * CC * PP;
  const float* kn = kk + (size_t)n * CC * PP;

  // ---- per-batch geometry (uniform across block) ----
  const float qx0 = cq[n*6+0], qy0 = cq[n*6+1], qz0 = cq[n*6+2];
  const float bwq = (cq[n*6+3] - qx0) * (1.0f/WW);
  const float bhq = (cq[n*6+4] - qy0) * (1.0f/HH);
  const float bdq = (cq[n*6+5] - qz0) * (1.0f/DD);
  const float kx0 = ck[n*6+0], ky0 = ck[n*6+1], kz0 = ck[n*6+2];
  const float bwk = (ck[n*6+3] - kx0) * (1.0f/WW);
  const float bhk = (ck[n*6+4] - ky0) * (1.0f/HH);
  const float bdk = (ck[n*6+5] - kz0) * (1.0f/DD);
  const float diagq = sqrtf(bwq*bwq + bhq*bhq + bdq*bdq);
  const float diagk = sqrtf(bwk*bwk + bhk*bhk + bdk*bdk);
  const float md    = fmaxf(diagq, diagk);
  const float thr2  = 0.25f * md * md;   // (0.5 * max_diag)^2

  // ---- row (q) centers for the 8 accumulator registers ----
  // C/D layout: VGPR r holds M=r (lanes 0-15) / M=r+8 (lanes 16-31).
  float rcx[8], rcy[8], rcz[8];
#pragma unroll
  for (int r = 0; r < 8; ++r) {
    const int pm = p0 + r + lhalf * 8;
    const int xm = pm & 15, ym = (pm >> 4) & 15, zm = pm >> 8;
    rcx[r] = ((float)xm + 0.5f) * bwq + qx0;
    rcy[r] = ((float)ym + 0.5f) * bhq + qy0;
    rcz[r] = ((float)zm + 0.5f) * bdq + qz0;
  }

  // ---- preload the A strip (q rows p0..p0+15, all 256 channels) into regs ----
  // A 16x4 f32 layout: M = lane&15 (both halves), K pair {0,1} (lanes 0-15)
  // or {2,3} (lanes 16-31). Per lane: only half the channels are needed.
  float A[128];
  {
    const float* ap = qn + (size_t)(lhalf * 2) * PP + (p0 + lmod);
#pragma unroll
    for (int it = 0; it < 64; ++it) {
      A[2*it]   = ap[(size_t)(4*it)     * PP];
      A[2*it+1] = ap[(size_t)(4*it + 1) * PP];
    }
  }

  __shared__ float Bs[2][CC * 16];   // double-buffered k tile, 2 x 16 KB
  __shared__ float redS[256];
  __shared__ float redC[256];

  const unsigned lds_base = (unsigned)(size_t)(&Bs[0][0]);  // low 32b = LDS offset

  float S = 0.0f, Cn = 0.0f;

  // ---- TDM prologue: stage first tile into buffer 0 ----
  if (wave == 0) {
    tdm_stage_ktile(kn, jt0 * 16, lds_base);
    __builtin_amdgcn_s_wait_tensorcnt(0);
  }
  __syncthreads();

  for (int t = 0; t < CT; ++t) {
    const int cur = t & 1;
    const int pj  = (jt0 + t) * 16;

    // prefetch next tile into the other buffer while computing this one
    if (wave == 0 && t + 1 < CT) {
      tdm_stage_ktile(kn, (jt0 + t + 1) * 16,
                      lds_base + (unsigned)((t + 1) & 1) * (CC * 16 * 4));
    }

    // ---- 16x16x256 tile via 64 x V_WMMA_F32_16X16X4_F32 ----
    const float* Bcur = &Bs[cur][0];
    v8f acc = {};
#pragma unroll
    for (int it = 0; it < 64; ++it) {
      const int k0 = it * 4;
      v2f a; a.x = A[2*it]; a.y = A[2*it+1];
      v2f bb;
      bb.x = Bcur[(k0 + lhalf * 2    ) * 16 + lmod];
      bb.y = Bcur[(k0 + lhalf * 2 + 1) * 16 + lmod];
      acc = __builtin_amdgcn_wmma_f32_16x16x4_f32(
          /*neg_a=*/false, a, /*neg_b=*/false, bb,
          /*c_mod=*/(short)0, acc, /*reuse_a=*/false, /*reuse_b=*/false);
    }

    // ---- mask + accumulate (column center depends on tile & lane only) ----
    const int pn = pj + lmod;
    const int xk = pn & 15, yk = (pn >> 4) & 15, zk = pn >> 8;
    const float ccx = ((float)xk + 0.5f) * bwk + kx0;
    const float ccy = ((float)yk + 0.5f) * bhk + ky0;
    const float ccz = ((float)zk + 0.5f) * bdk + kz0;
#pragma unroll
    for (int r = 0; r < 8; ++r) {
      const float dx = rcx[r] - ccx;
      const float dy = rcy[r] - ccy;
      const float dz = rcz[r] - ccz;
      const float d2 = dx*dx + dy*dy + dz*dz;
      if (d2 < thr2) { S += acc[r]; Cn += 1.0f; }
    }

    // publish the prefetched buffer: issuing wave drains TENSORcnt, then all
    // waves rendezvous (also protects the buffer we just finished reading).
    if (wave == 0) __builtin_amdgcn_s_wait_tensorcnt(0);
    __syncthreads();
  }

  // ---- deterministic block reduction ----
  redS[tid] = S;
  redC[tid] = Cn;
  __syncthreads();
  if (tid == 0) {
    float s = 0.0f, c = 0.0f;
    for (int i = 0; i < 256; ++i) { s += redS[i]; c += redC[i]; }
    partials[2*b]     = s;
    partials[2*b + 1] = c;
  }
}

__global__ void pixpro_final(const float* __restrict__ partials,
                             float* __restrict__ out) {
  if (threadIdx.x == 0 && blockIdx.x == 0) {
    float loss = 0.0f;
    for (int n = 0; n < NB; ++n) {
      float s = 0.0f, c = 0.0f;
      for (int r = 0; r < ROWG * COLS; ++r) {
        s += partials[2*(n*ROWG*COLS + r)];
        c += partials[2*(n*ROWG*COLS + r) + 1];
      }
      loss += s / (c + EPSF);
    }
    out[0] = -2.0f * loss / (float)NB;
  }
}

extern "C" void kernel_launch(void* const* d_in, const int* in_sizes, int n_in,
                              void* d_out, int out_size, void* d_ws, size_t ws_size,
                              hipStream_t stream) {
  const float* q  = (const float*)d_in[0];
  const float* k  = (const float*)d_in[1];
  const float* cq = (const float*)d_in[2];
  const float* ck = (const float*)d_in[3];
  float* partials = (float*)d_ws;     // 1024 blocks * 2 floats = 8 KB
  float* out      = (float*)d_out;

  pixpro_main<<<dim3(NB * ROWG * COLS), dim3(256), 0, stream>>>(q, k, cq, ck, partials);
  pixpro_final<<<dim3(1), dim3(1), 0, stream>>>(partials, out);
}